// EqPropNetwork_13580686590291
// MI455X (gfx1250) — compile-verified
//
#include <hip/hip_runtime.h>

typedef __attribute__((ext_vector_type(2))) float v2f;
typedef __attribute__((ext_vector_type(8))) float v8f;

#define B_ROWS   32768
#define IN_DIM   784
#define HID      512
#define OUT_DIM  10
#define T_STEPS  30
#define ROWS_PER_BLOCK 32
#define ROWS_PER_WAVE  16
#define H_STRIDE   516   // 512 + 4 pad: conflict-free C-tile & A-frag LDS access
#define W2T_STRIDE 520
#define OUT_STRIDE (HID + OUT_DIM)  // 522

__global__ __launch_bounds__(64) void eqprop_fused_kernel(
    const float* __restrict__ x, const float* __restrict__ h_init,
    const float* __restrict__ y_init, const float* __restrict__ W1,
    const float* __restrict__ b1, const float* __restrict__ W2,
    const float* __restrict__ b2, const long long* __restrict__ target,
    float* __restrict__ out)
{
    __shared__ float hS[ROWS_PER_BLOCK * H_STRIDE];   // 66048 B: h state
    __shared__ float w2t[12 * W2T_STRIDE];            // 24960 B: W2^T, K padded to 12
    __shared__ float w2p[512 * 16];                   // 32768 B: W2, N padded to 16
    __shared__ float yS[ROWS_PER_BLOCK * 16];         //  2048 B: y state, K padded

    const int tid    = threadIdx.x;
    const int lane   = tid & 31;
    const int wave   = tid >> 5;
    const int col    = lane & 15;
    const int khalf  = lane >> 4;     // 0: lanes 0-15, 1: lanes 16-31
    const int sel    = khalf * 2;     // K-slot select for 16x16x4 A/B frags
    const int blockRow0 = blockIdx.x * ROWS_PER_BLOCK;
    const int waveRowL  = wave * ROWS_PER_WAVE;       // local row base of this wave
    const int rowA_l    = waveRowL + col;             // local row for A-frags
    const int rowA_g    = blockRow0 + rowA_l;

    // ---------------- Phase 0: stage W2 / W2^T / y / h into LDS ----------------
    for (int i = tid; i < 512 * 16; i += blockDim.x) {
        int k = i >> 4, n = i & 15;
        w2p[i] = (n < OUT_DIM) ? W2[k * OUT_DIM + n] : 0.0f;
    }
    for (int i = tid; i < 12 * W2T_STRIDE; i += blockDim.x) {
        int k = i / W2T_STRIDE, n = i % W2T_STRIDE;
        float v = 0.0f;
        if (k < OUT_DIM && n < HID) v = W2[n * OUT_DIM + k];  // W2^T[k][n]
        w2t[i] = v;
    }
    for (int i = tid; i < ROWS_PER_BLOCK * 16; i += blockDim.x) {
        int r = i >> 4, n = i & 15;
        yS[i] = (n < OUT_DIM) ? y_init[(size_t)(blockRow0 + r) * OUT_DIM + n] : 0.0f;
    }
    {
        const float4* hg = (const float4*)(h_init + (size_t)blockRow0 * HID);
        for (int i = tid; i < ROWS_PER_BLOCK * (HID / 4); i += blockDim.x) {
            int r = i / (HID / 4), c4 = i % (HID / 4);
            *(float4*)&hS[r * H_STRIDE + c4 * 4] = hg[r * (HID / 4) + c4];
        }
    }
    __syncthreads();

    // Per-lane constant C-term for y-update: b2[col] + 0.5*onehot(row,col)
    float c0[8];
    {
        float b2v = (col < OUT_DIM) ? b2[col] : 0.0f;
#pragma unroll
        for (int r = 0; r < 8; ++r) {
            int rg = blockRow0 + waveRowL + r + 8 * khalf;
            long long tg = target[rg];
            c0[r] = b2v + ((tg == (long long)col) ? 0.5f : 0.0f);
        }
    }

    // ---------------- Phase A: c_h = x @ W1 + b1 (kept in VGPRs, C-layout) ----
    v8f acc[32];
#pragma unroll
    for (int t = 0; t < 32; ++t) {
        float bv = b1[t * 16 + col];
#pragma unroll
        for (int r = 0; r < 8; ++r) acc[t][r] = bv;
    }
    {
        const float* xp  = x  + (size_t)rowA_g * IN_DIM + sel;   // A: lane row, K=kk+sel
        const float* w1p = W1 + (size_t)sel * HID + col;         // B: K=kk+sel, N=t*16+col
#pragma unroll 1
        for (int kk = 0; kk < IN_DIM; kk += 4) {
            v2f a = *(const v2f*)(xp + kk);                      // {K=kk+sel, kk+sel+1}
#pragma unroll
            for (int t = 0; t < 32; ++t) {
                v2f b;
                b.x = w1p[(size_t)kk * HID + t * 16];
                b.y = w1p[(size_t)(kk + 1) * HID + t * 16];
                acc[t] = __builtin_amdgcn_wmma_f32_16x16x4_f32(
                    false, a, false, b, (short)0, acc[t], false, false);
            }
        }
    }

    // ---------------- Relaxation loop (Jacobi order preserved) ----------------
#pragma unroll 1
    for (int it = 0; it < T_STEPS; ++it) {
        // (1) accy = h_old @ W2 + (b2 + 0.5*onehot + 0.5*y_old)   [reads h_old]
        // K=512 reduction split 4-way to break the WMMA D->C dependency chain.
        v8f accy0, accy1, accy2, accy3;
#pragma unroll
        for (int r = 0; r < 8; ++r) {
            accy0[r] = c0[r] + 0.5f * yS[(waveRowL + r + 8 * khalf) * 16 + col];
            accy1[r] = 0.0f;
            accy2[r] = 0.0f;
            accy3[r] = 0.0f;
        }
        {
            const float* hA = &hS[rowA_l * H_STRIDE + sel];
            const float* wB = &w2p[sel * 16 + col];
#pragma unroll 1
            for (int kk = 0; kk < HID; kk += 16) {
                v2f a0 = *(const v2f*)(hA + kk);
                v2f a1 = *(const v2f*)(hA + kk + 4);
                v2f a2 = *(const v2f*)(hA + kk + 8);
                v2f a3 = *(const v2f*)(hA + kk + 12);
                v2f b0, b1_, b2_, b3_;
                b0.x  = wB[(kk +  0) * 16];  b0.y  = wB[(kk +  1) * 16];
                b1_.x = wB[(kk +  4) * 16];  b1_.y = wB[(kk +  5) * 16];
                b2_.x = wB[(kk +  8) * 16];  b2_.y = wB[(kk +  9) * 16];
                b3_.x = wB[(kk + 12) * 16];  b3_.y = wB[(kk + 13) * 16];
                accy0 = __builtin_amdgcn_wmma_f32_16x16x4_f32(
                    false, a0, false, b0, (short)0, accy0, false, false);
                accy1 = __builtin_amdgcn_wmma_f32_16x16x4_f32(
                    false, a1, false, b1_, (short)0, accy1, false, false);
                accy2 = __builtin_amdgcn_wmma_f32_16x16x4_f32(
                    false, a2, false, b2_, (short)0, accy2, false, false);
                accy3 = __builtin_amdgcn_wmma_f32_16x16x4_f32(
                    false, a3, false, b3_, (short)0, accy3, false, false);
            }
        }

        // (2) h_new = clip(0.5*(h_old + c_h + y_old @ W2^T))      [reads y_old, writes h]
        {
            const float* yA = &yS[rowA_l * 16 + sel];
            v2f ya0 = *(const v2f*)(yA);        // K = sel..sel+1
            v2f ya1 = *(const v2f*)(yA + 4);    // K = 4+sel..
            v2f ya2 = *(const v2f*)(yA + 8);    // K = 8+sel.. (cols 10,11 are zero pad)
            const float* wT = &w2t[sel * W2T_STRIDE + col];
#pragma unroll
            for (int t = 0; t < 32; ++t) {
                v2f wb0, wb1, wb2;
                wb0.x = wT[t * 16];
                wb0.y = wT[1 * W2T_STRIDE + t * 16];
                wb1.x = wT[4 * W2T_STRIDE + t * 16];
                wb1.y = wT[5 * W2T_STRIDE + t * 16];
                wb2.x = wT[8 * W2T_STRIDE + t * 16];
                wb2.y = wT[9 * W2T_STRIDE + t * 16];
                v8f s = __builtin_amdgcn_wmma_f32_16x16x4_f32(
                    false, ya0, false, wb0, (short)0, acc[t], false, false);
                s = __builtin_amdgcn_wmma_f32_16x16x4_f32(
                    false, ya1, false, wb1, (short)0, s, false, false);
                s = __builtin_amdgcn_wmma_f32_16x16x4_f32(
                    false, ya2, false, wb2, (short)0, s, false, false);
#pragma unroll
                for (int r = 0; r < 8; ++r) {
                    int la = (waveRowL + r + 8 * khalf) * H_STRIDE + t * 16 + col;
                    float hn = 0.5f * (hS[la] + s[r]);
                    hn = fminf(fmaxf(hn, 0.0f), 1.0f);
                    hS[la] = hn;
                }
            }
        }

        // (3) y_new = clip(0.5 * sum of partial accumulators)     [writes y]
#pragma unroll
        for (int r = 0; r < 8; ++r) {
            float yn = 0.5f * ((accy0[r] + accy1[r]) + (accy2[r] + accy3[r]));
            yn = fminf(fmaxf(yn, 0.0f), 1.0f);
            int la = (waveRowL + r + 8 * khalf) * 16 + col;
            yS[la] = (col < OUT_DIM) ? yn : 0.0f;   // keep K-padding zero
        }
    }

    __syncthreads();

    // ---------------- Output: concat(h, y) -> [B, 522] ----------------
    {
        float* og = out + (size_t)blockRow0 * OUT_STRIDE;
        for (int i = tid; i < ROWS_PER_BLOCK * HID; i += blockDim.x) {
            int r = i / HID, c = i % HID;
            og[(size_t)r * OUT_STRIDE + c] = hS[r * H_STRIDE + c];
        }
        for (int i = tid; i < ROWS_PER_BLOCK * OUT_DIM; i += blockDim.x) {
            int r = i / OUT_DIM, c = i % OUT_DIM;
            og[(size_t)r * OUT_STRIDE + HID + c] = yS[r * 16 + c];
        }
    }
}

extern "C" void kernel_launch(void* const* d_in, const int* in_sizes, int n_in,
                              void* d_out, int out_size, void* d_ws, size_t ws_size,
                              hipStream_t stream) {
    (void)in_sizes; (void)n_in; (void)out_size; (void)d_ws; (void)ws_size;
    const float*     x      = (const float*)d_in[0];
    const float*     h_init = (const float*)d_in[1];
    const float*     y_init = (const float*)d_in[2];
    const float*     W1     = (const float*)d_in[3];
    const float*     b1     = (const float*)d_in[4];
    const float*     W2     = (const float*)d_in[5];
    const float*     b2     = (const float*)d_in[6];
    const long long* target = (const long long*)d_in[7];
    float* out = (float*)d_out;

    dim3 grid(B_ROWS / ROWS_PER_BLOCK);   // 1024 blocks
    dim3 block(64);                       // 2 waves, 16 rows each
    eqprop_fused_kernel<<<grid, block, 0, stream>>>(
        x, h_init, y_init, W1, b1, W2, b2, target, out);
}